// GIN_66365834658285
// MI455X (gfx1250) — compile-verified
//
#include <hip/hip_runtime.h>

typedef __attribute__((ext_vector_type(2))) float v2f;
typedef __attribute__((ext_vector_type(8))) float v8f;

#define NN_NODES 100000
#define NN_EDGES 1600000
#define DH 64
#define NGR 64
#define STRIPS 8   // 16-row strips processed per wave in the K=64 GEMM

// ---------------- elementwise helpers ----------------
__global__ void gin_copy_f4(float4* __restrict__ dst, const float4* __restrict__ src, int n4) {
    int i = blockIdx.x * blockDim.x + threadIdx.x;
    if (i < n4) dst[i] = src[i];
}

__global__ void gin_zero_f(float* __restrict__ p, int n) {
    int i = blockIdx.x * blockDim.x + threadIdx.x;
    if (i < n) p[i] = 0.0f;
}

// ---------------- edge scatter: agg[dst] += h[src] ----------------
__global__ void gin_scatter(float* __restrict__ agg, const float* __restrict__ h,
                            const int* __restrict__ src, const int* __restrict__ dst, int nE) {
    int tid = blockIdx.x * blockDim.x + threadIdx.x;
    int e = tid >> 4;
    if (e >= nE) return;
    int g = (tid & 15) << 2;
    int s = src[e], d = dst[e];
    float4 v = *(const float4*)(h + (size_t)s * DH + g);
    float* p = agg + (size_t)d * DH + g;
    atomicAdd(p + 0, v.x);
    atomicAdd(p + 1, v.y);
    atomicAdd(p + 2, v.z);
    atomicAdd(p + 3, v.w);
}

// ---------------- K=64 WMMA GEMM with register-resident B ----------------
// D = A[M,64] @ W[64,N] + bias (opt relu). M%16==0, N%16==0.
// Each wave owns one 16-col tile; preloads all 16 B fragments (32 VGPRs),
// then sweeps STRIPS row strips issuing 16 v_wmma_f32_16x16x4_f32 each.
__global__ void gin_gemm_k64(const float* __restrict__ A, const float* __restrict__ W,
                             const float* __restrict__ bias, float* __restrict__ D,
                             int M, int N, int relu) {
    const int lane = threadIdx.x & 31;
    const int wave = threadIdx.x >> 5;
    const int tilesN = N >> 4;
    const int gid = blockIdx.x * (blockDim.x >> 5) + wave;
    const int tn = gid % tilesN;
    const int rowGroup = gid / tilesN;
    const int stripsTotal = M >> 4;
    const int tm0 = rowGroup * STRIPS;
    if (tm0 >= stripsTotal) return;                 // wave-uniform
    const int half = lane >> 4, l16 = lane & 15;
    const int n = (tn << 4) + l16;

    // Preload B fragments (K=64 -> 16 fragments), reused for every row strip.
    v2f bfrag[16];
#pragma unroll
    for (int kk = 0; kk < 16; ++kk) {
        const int ka = (kk << 2) + (half << 1);
        bfrag[kk].x = W[(size_t)ka * N + n];
        bfrag[kk].y = W[(size_t)(ka + 1) * N + n];
    }
    const float bv = bias[n];

    const int nStrips = min(STRIPS, stripsTotal - tm0);   // wave-uniform
    const float* aptr = A + ((size_t)(tm0 << 4) + l16) * DH + (half << 1);
    for (int s = 0; s < nStrips; ++s) {
        __builtin_prefetch(aptr + 16 * DH, 0, 3);   // global_prefetch_b8: next strip's A row
        v8f acc = {};
#pragma unroll
        for (int kk = 0; kk < 16; ++kk) {
            v2f a = *(const v2f*)(aptr + (kk << 2));
            acc = __builtin_amdgcn_wmma_f32_16x16x4_f32(
                false, a, false, bfrag[kk], (short)0, acc, false, false);
        }
        const int row0 = (tm0 + s) << 4;
#pragma unroll
        for (int r = 0; r < 8; ++r) {
            float v = acc[r] + bv;
            if (relu) v = fmaxf(v, 0.0f);
            D[(size_t)(row0 + r + (half << 3)) * N + n] = v;
        }
        aptr += (size_t)16 * DH;
    }
}

// ---------------- generic WMMA GEMM (runtime K) for the K=256 MLP stage ----------------
__global__ void gin_gemm_wmma(const float* __restrict__ A, const float* __restrict__ W,
                              const float* __restrict__ bias, float* __restrict__ D,
                              int M, int K, int N, int relu) {
    const int lane = threadIdx.x & 31;
    const int wave = threadIdx.x >> 5;
    const int tilesN = N >> 4;
    const int tile = blockIdx.x * (blockDim.x >> 5) + wave;
    if (tile >= (M >> 4) * tilesN) return;
    const int tm = tile / tilesN, tn = tile % tilesN;
    const int row0 = tm << 4, col0 = tn << 4;
    const int half = lane >> 4, l16 = lane & 15;
    const int m = row0 + l16;
    const int n = col0 + l16;
    v8f acc = {};
    const float* arow = A + (size_t)m * K + (half << 1);
    for (int k = 0; k < K; k += 4) {
        v2f a = *(const v2f*)(arow + k);
        const int ka = k + (half << 1);
        v2f b;
        b.x = W[(size_t)ka * N + n];
        b.y = W[(size_t)(ka + 1) * N + n];
        acc = __builtin_amdgcn_wmma_f32_16x16x4_f32(
            false, a, false, b, (short)0, acc, false, false);
    }
    const float bv = bias[n];
#pragma unroll
    for (int r = 0; r < 8; ++r) {
        float v = acc[r] + bv;
        if (relu) v = fmaxf(v, 0.0f);
        D[(size_t)(row0 + r + (half << 3)) * N + n] = v;
    }
}

// ---------------- BatchNorm (training-mode batch stats) ----------------
__global__ void gin_bn_stats(const float* __restrict__ h, float* __restrict__ sum,
                             float* __restrict__ sumsq, int M) {
    int tid = blockIdx.x * blockDim.x + threadIdx.x;
    int c = tid & (DH - 1);
    int r0 = tid >> 6;
    int rstride = (gridDim.x * blockDim.x) >> 6;
    float s = 0.0f, s2 = 0.0f;
    for (int r = r0; r < M; r += rstride) {
        float v = h[(size_t)r * DH + c];
        s += v;
        s2 += v * v;
    }
    atomicAdd(sum + c, s);
    atomicAdd(sumsq + c, s2);
}

__global__ void gin_bn_finalize(const float* __restrict__ sum, const float* __restrict__ sumsq,
                                const float* __restrict__ gamma, const float* __restrict__ beta,
                                float* __restrict__ scale, float* __restrict__ shift, float invM) {
    int c = threadIdx.x;
    if (c < DH) {
        float mean = sum[c] * invM;
        float var = sumsq[c] * invM - mean * mean;
        float sc = gamma[c] * rsqrtf(var + 1e-5f);
        scale[c] = sc;
        shift[c] = beta[c] - mean * sc;
    }
}

__global__ void gin_bn_apply_relu(const float* __restrict__ in, const float* __restrict__ scale,
                                  const float* __restrict__ shift, float* __restrict__ out, int n) {
    int i = blockIdx.x * blockDim.x + threadIdx.x;
    if (i >= n) return;
    int c = i & (DH - 1);
    out[i] = fmaxf(fmaf(in[i], scale[c], shift[c]), 0.0f);
}

// ---------------- global add pool ----------------
__global__ void gin_pool(const float* __restrict__ h, const int* __restrict__ batch,
                         float* __restrict__ pooled, int M) {
    int tid = blockIdx.x * blockDim.x + threadIdx.x;
    int node = tid >> 4;
    if (node >= M) return;
    int g4 = (tid & 15) << 2;
    int g = batch[node];
    float4 v = *(const float4*)(h + (size_t)node * DH + g4);
    float* p = pooled + (size_t)g * DH + g4;
    atomicAdd(p + 0, v.x);
    atomicAdd(p + 1, v.y);
    atomicAdd(p + 2, v.z);
    atomicAdd(p + 3, v.w);
}

static inline int cdiv(int a, int b) { return (a + b - 1) / b; }

extern "C" void kernel_launch(void* const* d_in, const int* in_sizes, int n_in,
                              void* d_out, int out_size, void* d_ws, size_t ws_size,
                              hipStream_t stream) {
    (void)in_sizes; (void)n_in; (void)out_size; (void)ws_size;

    const float* x     = (const float*)d_in[0];
    const int*   esrc  = (const int*)d_in[1];
    const int*   edst  = esrc + NN_EDGES;
    const int*   batch = (const int*)d_in[2];
    const float* cw1[3] = {(const float*)d_in[3],  (const float*)d_in[7],  (const float*)d_in[11]};
    const float* cb1[3] = {(const float*)d_in[4],  (const float*)d_in[8],  (const float*)d_in[12]};
    const float* cw2[3] = {(const float*)d_in[5],  (const float*)d_in[9],  (const float*)d_in[13]};
    const float* cb2[3] = {(const float*)d_in[6],  (const float*)d_in[10], (const float*)d_in[14]};
    const float* bng[2] = {(const float*)d_in[15], (const float*)d_in[17]};
    const float* bnb[2] = {(const float*)d_in[16], (const float*)d_in[18]};
    const float* mp_w1 = (const float*)d_in[19];
    const float* mp_b1 = (const float*)d_in[20];
    const float* mp_w2 = (const float*)d_in[21];
    const float* mp_b2 = (const float*)d_in[22];

    const size_t NF = (size_t)NN_NODES * DH;   // 6,400,000 floats
    float* bufA  = (float*)d_ws;
    float* bufB  = bufA + NF;
    float* bufC  = bufB + NF;
    float* stats = bufC + NF;                  // sum|sumsq|scale|shift : 4*64 floats
    float* hid   = stats + 256;                // 64*256 floats

    float* outp   = (float*)d_out;             // [64,128]
    float* pooled = outp + (size_t)NGR * 128;  // emb region [64,64] doubles as pool accumulator

    const int n4       = (int)(NF / 4);                        // 1,600,000
    const int copyBlks = cdiv(n4, 256);                        // 6250
    const int scatBlks = cdiv(NN_EDGES * 16, 256);             // 100000
    const int ewBlks   = cdiv((int)NF, 256);                   // 25000
    const int poolBlks = cdiv(NN_NODES * 16, 256);             // 6250
    const float invM   = 1.0f / (float)NN_NODES;

    // K=64 GEMM launch geometry: waves = ceil(strips/STRIPS) * tilesN
    const int nodeStrips = NN_NODES / 16;                      // 6250
    const int nodeWaves  = cdiv(nodeStrips, STRIPS) * (DH / 16);   // 782*4 = 3128
    const int nodeBlks   = cdiv(nodeWaves, 8);                 // 391

    // layer buffer rotation: h -> agg -> t1 -> t2 -> h'
    const float* h = x;
    float* rot[3] = {bufA, bufB, bufC};
    for (int l = 0; l < 3; ++l) {
        float* agg = rot[0];
        float* t1  = rot[1];
        float* t2  = rot[2];

        // agg = h + sum_{j->i} h_j
        gin_copy_f4<<<copyBlks, 256, 0, stream>>>((float4*)agg, (const float4*)h, n4);
        gin_scatter<<<scatBlks, 256, 0, stream>>>(agg, h, esrc, edst, NN_EDGES);
        // t1 = relu(agg @ w1 + b1) ; t2 = t1 @ w2 + b2
        gin_gemm_k64<<<nodeBlks, 256, 0, stream>>>(agg, cw1[l], cb1[l], t1, NN_NODES, DH, 1);
        gin_gemm_k64<<<nodeBlks, 256, 0, stream>>>(t1, cw2[l], cb2[l], t2, NN_NODES, DH, 0);
        if (l < 2) {
            gin_zero_f<<<1, 256, 0, stream>>>(stats, 128);
            gin_bn_stats<<<64, 256, 0, stream>>>(t2, stats, stats + 64, NN_NODES);
            gin_bn_finalize<<<1, 64, 0, stream>>>(stats, stats + 64, bng[l], bnb[l],
                                                  stats + 128, stats + 192, invM);
            gin_bn_apply_relu<<<ewBlks, 256, 0, stream>>>(t2, stats + 128, stats + 192,
                                                          agg, (int)NF);
            h = agg;
            float* nrot[3] = {t1, t2, rot[0]};
            rot[0] = nrot[0]; rot[1] = nrot[1]; rot[2] = nrot[2];
        } else {
            h = t2;
        }
    }

    // emb = global add pool (accumulated directly into d_out's emb region)
    gin_zero_f<<<cdiv(NGR * DH, 256), 256, 0, stream>>>(pooled, NGR * DH);
    gin_pool<<<poolBlks, 256, 0, stream>>>(h, batch, pooled, NN_NODES);

    // out = relu(pooled @ mp_w1 + mp_b1) @ mp_w2 + mp_b2
    //   MLP stage 1 is K=64 -> reuse the register-resident-B kernel (4 strips, 16 col tiles)
    {
        int strips = NGR / 16;                                         // 4
        int waves  = cdiv(strips, STRIPS) * (256 / 16);                // 1*16
        gin_gemm_k64<<<cdiv(waves, 8), 256, 0, stream>>>(pooled, mp_w1, mp_b1, hid,
                                                         NGR, 256, 1);
    }
    gin_gemm_wmma<<<cdiv((NGR / 16) * (128 / 16), 8), 256, 0, stream>>>(
        hid, mp_w2, mp_b2, outp, NGR, 256, 128, 0);
}